// EdgePointCloudGenerator_48601849921595
// MI455X (gfx1250) — compile-verified
//
#include <hip/hip_runtime.h>
#include <math.h>

// Problem constants (from reference)
#define BATCH 32
#define CCH   32
#define RESO  16
#define GCELLS 4096     // RESO^3
#define NPTS  16384
#define HDIM  64
#define KIN   34        // C + 2

typedef __attribute__((ext_vector_type(2))) float v2f;
typedef __attribute__((ext_vector_type(8))) float v8f;

// ---------------------------------------------------------------------------
// Phase 1: per-batch cumsum(counts) inverted into slot->cell index map.
// searchsorted(cum, j, 'right') == g  <=>  j in [cum[g]-counts[g], cum[g])
// One block per batch; 256 threads, 16 cells each.
// ---------------------------------------------------------------------------
__global__ void epcg_scatter_idx(const int* __restrict__ counts,
                                 int* __restrict__ idx) {
  __shared__ int s[256];
  const int b = blockIdx.x;
  const int t = threadIdx.x;
  const int* cb = counts + b * GCELLS;
  int local[16];
  int sum = 0;
#pragma unroll
  for (int i = 0; i < 16; ++i) { local[i] = cb[t * 16 + i]; sum += local[i]; }
  s[t] = sum;
  __syncthreads();
  if (t == 0) {               // tiny serial exclusive scan (256 adds)
    int run = 0;
    for (int i = 0; i < 256; ++i) { int v = s[i]; s[i] = run; run += v; }
  }
  __syncthreads();
  int start = s[t];
  int* ib = idx + b * NPTS;
#pragma unroll
  for (int i = 0; i < 16; ++i) {
    const int g = t * 16 + i;
    const int c = local[i];
    for (int q = 0; q < c; ++q) ib[start + q] = g;
    start += c;
  }
}

// ---------------------------------------------------------------------------
// Phase 2: gather + MLP with V_WMMA_F32_16X16X4_F32 (full fp32 precision).
// One wave handles one 16-point tile per loop iteration:
//   layer1: 4 M-tiles x 9 K-steps of 16x16x4 f32 WMMA (b1 folded into K pad)
//   layer2: VALU dot (3x64) from accumulator layout + shfl_xor lane-half fold
// ---------------------------------------------------------------------------
__global__ void __launch_bounds__(256)
epcg_mlp_wmma(const float* __restrict__ x,       // (B, C, G)
              const float* __restrict__ b_rnd,   // (B, 2, N)
              const float* __restrict__ grid_o,  // (3, G)
              const float* __restrict__ W1,      // (64, 34)
              const float* __restrict__ b1,      // (64)
              const float* __restrict__ W2,      // (3, 64)
              const float* __restrict__ b2,      // (3)
              const int*   __restrict__ idx,     // (B, N)
              float* __restrict__ out,           // (B, 3, N)
              float* __restrict__ reg) {         // (B, N)
  // Stage W2 transposed+padded into LDS: w2lds[k*4 + o], 16B-aligned rows.
  __shared__ float w2lds[HDIM * 4];
  const int tid = threadIdx.x;
  if (tid < 192) {
    const int o = tid >> 6, k = tid & 63;
    w2lds[k * 4 + o] = W2[tid];
  }
  __syncthreads();

  const int lane = tid & 31;
  const int hi   = lane >> 4;       // lane-half: selects K-subcolumns
  const int m    = lane & 15;       // row (A) / point-in-tile (B,D)
  const int wave   = blockIdx.x * (blockDim.x >> 5) + (tid >> 5);
  const int nwaves = gridDim.x * (blockDim.x >> 5);

  // --- Preload A fragments (W1, with b1 folded into the K=8 pad step) ------
  // A layout (16x4 f32): lane<16 -> cols (4k,4k+1); lane>=16 -> (4k+2,4k+3).
  v2f afrag[4][9];
#pragma unroll
  for (int t = 0; t < 4; ++t) {
    const int row = 16 * t + m;
    const float* wrow = W1 + row * KIN;
#pragma unroll
    for (int kk = 0; kk < 8; ++kk) {
      const int c0 = 4 * kk + 2 * hi;       // even -> 8B-aligned v2f load
      afrag[t][kk] = *(const v2f*)(wrow + c0);
    }
    if (!hi) {
      afrag[t][8] = *(const v2f*)(wrow + 32);     // real cols 32,33
    } else {
      v2f a; a.x = b1[row]; a.y = 0.0f;           // bias column (pairs B=1.0)
      afrag[t][8] = a;
    }
  }
  const float b2v0 = b2[0], b2v1 = b2[1], b2v2 = b2[2];
  const v8f vzero = {0.f, 0.f, 0.f, 0.f, 0.f, 0.f, 0.f, 0.f};

  const int tilesTotal = BATCH * (NPTS / 16);      // 32768
  for (int tile = wave; tile < tilesTotal; tile += nwaves) {
    const int b   = tile >> 10;                    // tile / (N/16)
    const int tin = tile & 1023;
    const int j   = tin * 16 + m;                  // point id in batch
    const int cell = idx[b * NPTS + j];
    const float* xb = x + (size_t)b * CCH * GCELLS + cell;

    v8f acc[4];
#pragma unroll
    for (int t = 0; t < 4; ++t) acc[t] = vzero;

    // --- Layer 1: K-steps 0..7 gather x channels, step 8 = b_rnd + bias ---
#pragma unroll
    for (int kk = 0; kk < 8; ++kk) {
      const int c0 = 4 * kk + 2 * hi;
      v2f bfrag;
      bfrag.x = xb[(size_t)c0 * GCELLS];
      bfrag.y = xb[(size_t)(c0 + 1) * GCELLS];
#pragma unroll
      for (int t = 0; t < 4; ++t)
        acc[t] = __builtin_amdgcn_wmma_f32_16x16x4_f32(
            false, afrag[t][kk], false, bfrag, (short)0, acc[t], false, false);
    }
    {
      v2f bfrag;
      if (!hi) {  // rows 32,33 = Halton noise
        const size_t rb = (size_t)b * 2 * NPTS + j;
        bfrag.x = b_rnd[rb];
        bfrag.y = b_rnd[rb + NPTS];
      } else {    // rows 34,35 = (1, 0): multiplies the folded b1 column
        bfrag.x = 1.0f; bfrag.y = 0.0f;
      }
#pragma unroll
      for (int t = 0; t < 4; ++t)
        acc[t] = __builtin_amdgcn_wmma_f32_16x16x4_f32(
            false, afrag[t][8], false, bfrag, (short)0, acc[t], false, false);
    }

    // --- Layer 2: ReLU + 3x64 dot. Lane holds h rows 16t+r+8*hi of point m.
    float o0 = 0.f, o1 = 0.f, o2 = 0.f;
#pragma unroll
    for (int t = 0; t < 4; ++t) {
#pragma unroll
      for (int r = 0; r < 8; ++r) {
        float h = acc[t][r];
        h = h > 0.f ? h : 0.f;
        const int k = 16 * t + r + 8 * hi;
        const float* w = &w2lds[k * 4];            // ds_load_b128-friendly
        o0 = fmaf(w[0], h, o0);
        o1 = fmaf(w[1], h, o1);
        o2 = fmaf(w[2], h, o2);
      }
    }
    // Fold the two lane-halves (rows +8) together.
    o0 += __shfl_xor(o0, 16, 32);
    o1 += __shfl_xor(o1, 16, 32);
    o2 += __shfl_xor(o2, 16, 32);

    if (!hi) {   // lanes 0..15 own one point each; store 3 outs + reg
      o0 += b2v0; o1 += b2v1; o2 += b2v2;
      const float nrm = sqrtf(o0 * o0 + o1 * o1 + o2 * o2);
      float rg = nrm - 0.10825317547305482f;       // sqrt(3)/RES
      rg = rg > 0.f ? rg : 0.f;
      const float g0 = grid_o[cell];
      const float g1 = grid_o[GCELLS + cell];
      const float g2 = grid_o[2 * GCELLS + cell];
      const size_t ob = (size_t)b * 3 * NPTS + j;
      out[ob]            = o0 + g0;
      out[ob + NPTS]     = o1 + g1;
      out[ob + 2 * NPTS] = o2 + g2;
      reg[(size_t)b * NPTS + j] = rg;
    }
  }
}

// ---------------------------------------------------------------------------
extern "C" void kernel_launch(void* const* d_in, const int* in_sizes, int n_in,
                              void* d_out, int out_size, void* d_ws, size_t ws_size,
                              hipStream_t stream) {
  const float* x      = (const float*)d_in[0];   // (B,C,16,16,16)
  const int*   counts = (const int*)  d_in[1];   // (B,G)
  const float* b_rnd  = (const float*)d_in[2];   // (B,2,N)
  const float* grid_o = (const float*)d_in[3];   // (3,G)
  const float* W1     = (const float*)d_in[4];   // (64,34)
  const float* b1v    = (const float*)d_in[5];   // (64)
  const float* W2     = (const float*)d_in[6];   // (3,64)
  const float* b2v    = (const float*)d_in[7];   // (3)

  int* idx = (int*)d_ws;                          // B*N ints = 2 MB scratch
  float* out = (float*)d_out;                     // (B,3,N)
  float* reg = out + (size_t)BATCH * 3 * NPTS;    // (B,N)

  epcg_scatter_idx<<<BATCH, 256, 0, stream>>>(counts, idx);
  // 256 blocks x 8 waves = 2048 waves; 16 tiles per wave (uniform trip count)
  epcg_mlp_wmma<<<256, 256, 0, stream>>>(x, b_rnd, grid_o, W1, b1v, W2, b2v,
                                         idx, out, reg);
}